// Word2Vec_86947317940479
// MI455X (gfx1250) — compile-verified
//
#include <hip/hip_runtime.h>

typedef float v2f __attribute__((ext_vector_type(2)));
typedef float v8f __attribute__((ext_vector_type(8)));

#define VOCAB  100000
#define EMB    128
#define BATCH  16384
#define PPOS   4
#define NNEG   20
#define NSAMP  24            // P + N
#define EPSL   1e-3f

#define WAVES_PER_BLOCK 8
#define NBLOCKS (BATCH / WAVES_PER_BLOCK)   // 2048

// One wave per batch element. 24x128 matvec done as two 16x16x4-f32 WMMA
// accumulation chains (K looped in chunks of 4), B operand = embedding
// vector broadcast across the 16 N columns.
__global__ __launch_bounds__(256)
void w2v_loss_kernel(const int*   __restrict__ x,
                     const int*   __restrict__ pos,
                     const int*   __restrict__ neg,
                     const float* __restrict__ emb_table,
                     const float* __restrict__ out_weight,
                     float*       __restrict__ partial)
{
    const int lane = threadIdx.x & 31;
    const int wave = threadIdx.x >> 5;
    const int b    = blockIdx.x * WAVES_PER_BLOCK + wave;

    const int r    = lane & 15;     // A-matrix row within tile
    const int half = lane >> 4;     // 0: holds K={0,1}; 1: holds K={2,3}
    const int col  = half * 2;      // float column offset within 4-wide K chunk

    // Sample index for a logical sample slot s (0..3 positive, 4..23 negative)
    auto sample_idx = [&](int s) -> int {
        return (s < PPOS) ? pos[b * PPOS + s] : neg[b * NNEG + (s - PPOS)];
    };

    const int s0  = r;                                   // tile0 rows: samples 0..15
    const int s1t = 16 + r;                              // tile1 rows: samples 16..31
    const int s1  = (s1t < NSAMP) ? s1t : (NSAMP - 1);   // clamp; padded rows' outputs ignored

    const int i0 = sample_idx(s0);
    const int i1 = sample_idx(s1);
    const int xb = x[b];

    const float* ap0 = out_weight + (size_t)i0 * EMB + col;
    const float* ap1 = out_weight + (size_t)i1 * EMB + col;
    const float* bp  = emb_table  + (size_t)xb * EMB + col;

    v8f c0 = {};
    v8f c1 = {};

#pragma unroll
    for (int kc = 0; kc < EMB / 4; ++kc) {
        v2f a0 = *(const v2f*)(ap0 + kc * 4);
        v2f a1 = *(const v2f*)(ap1 + kc * 4);
        v2f bb = *(const v2f*)(bp  + kc * 4);
        // (neg_a, A, neg_b, B, c_mod, C, reuse_a, reuse_b)
        c0 = __builtin_amdgcn_wmma_f32_16x16x4_f32(false, a0, false, bb,
                                                   (short)0, c0, false, false);
        c1 = __builtin_amdgcn_wmma_f32_16x16x4_f32(false, a1, false, bb,
                                                   (short)0, c1, false, false);
    }

    // D layout: lanes 0-15 hold M=j in c[j]; lanes 16-31 hold M=j+8 in c[j].
    // All N columns identical (B broadcast) -> read from lanes 0 and 16 only.
    float psum = 0.0f;
    float nsum = 0.0f;
    if (r == 0) {
#pragma unroll
        for (int j = 0; j < 8; ++j) {
            const int   s     = j + 8 * half;   // tile0 sample id 0..15
            const float logit = c0[j];
            if (s < PPOS) {
                const float p = 1.0f / (1.0f + expf(-logit));   // sigmoid(x)
                psum += -logf(p);
            } else {
                const float q = 1.0f / (1.0f + expf(logit));    // 1 - sigmoid(x)
                nsum += -logf(q + EPSL);
            }
        }
        if (half == 0) {
#pragma unroll
            for (int j = 0; j < 8; ++j) {       // tile1 samples 16..23 (all negative)
                const float logit = c1[j];
                const float q = 1.0f / (1.0f + expf(logit));
                nsum += -logf(q + EPSL);
            }
        }
    }

    // Wave reduction (only lanes 0 and 16 are nonzero)
#pragma unroll
    for (int off = 16; off > 0; off >>= 1) {
        psum += __shfl_xor(psum, off, 32);
        nsum += __shfl_xor(nsum, off, 32);
    }

    __shared__ float sp[WAVES_PER_BLOCK];
    __shared__ float sn[WAVES_PER_BLOCK];
    if (lane == 0) { sp[wave] = psum; sn[wave] = nsum; }
    __syncthreads();

    if (threadIdx.x == 0) {
        float P = 0.0f, Nn = 0.0f;
#pragma unroll
        for (int i = 0; i < WAVES_PER_BLOCK; ++i) { P += sp[i]; Nn += sn[i]; }
        partial[blockIdx.x * 2 + 0] = P;
        partial[blockIdx.x * 2 + 1] = Nn;
    }
}

// Deterministic single-block finalize: fixed-order strided accumulation +
// LDS tree reduction, then write the scalar loss.
__global__ __launch_bounds__(256)
void w2v_finalize(const float* __restrict__ partial, float* __restrict__ out)
{
    __shared__ float sp[256];
    __shared__ float sn[256];
    float P = 0.0f, Nn = 0.0f;
    for (int i = threadIdx.x; i < NBLOCKS; i += 256) {
        P  += partial[2 * i + 0];
        Nn += partial[2 * i + 1];
    }
    sp[threadIdx.x] = P;
    sn[threadIdx.x] = Nn;
    __syncthreads();
#pragma unroll
    for (int off = 128; off > 0; off >>= 1) {
        if ((int)threadIdx.x < off) {
            sp[threadIdx.x] += sp[threadIdx.x + off];
            sn[threadIdx.x] += sn[threadIdx.x + off];
        }
        __syncthreads();
    }
    if (threadIdx.x == 0) {
        out[0] = sp[0] / (float)(BATCH * PPOS) + sn[0] / (float)(BATCH * NNEG);
    }
}

extern "C" void kernel_launch(void* const* d_in, const int* in_sizes, int n_in,
                              void* d_out, int out_size, void* d_ws, size_t ws_size,
                              hipStream_t stream)
{
    (void)in_sizes; (void)n_in; (void)out_size; (void)ws_size;
    const int*   x   = (const int*)d_in[0];
    const int*   pos = (const int*)d_in[1];
    const int*   neg = (const int*)d_in[2];
    const float* emb = (const float*)d_in[3];
    const float* w   = (const float*)d_in[4];
    float* partial = (float*)d_ws;                 // 2 * NBLOCKS floats (16 KB)

    w2v_loss_kernel<<<NBLOCKS, 256, 0, stream>>>(x, pos, neg, emb, w, partial);
    w2v_finalize<<<1, 256, 0, stream>>>(partial, (float*)d_out);
}